// MVProdDense_60069412602337
// MI455X (gfx1250) — compile-verified
//
#include <hip/hip_runtime.h>

// ---------------------------------------------------------------------------
// GA dense layer on MI455X (gfx1250, wave32):
//   out[o] (128x64) = sum_{ia=0..7} sign(ia,o) * x[ia](128x128) @ w[ib(ia,o)](128x64)
// 64 f32 GEMMs, 134 MFLOP, 1MB unique data -> L2-resident, latency bound.
// V_WMMA_F32_16X16X4_F32 keeps full f32 precision (reference is f32).
//
// v3 changes vs v2:
//  - Cayley table fully arithmetic: blade permutation as a 32-bit nibble
//    immediate (0x76534210), sign via closed-form parity. The whole row for
//    this wave's o is packed once in the prologue into ibword (8x3 bits) and
//    negword (8x1 bit) -> no s_load/kmcnt waits in the inner loop.
//  - Rest unchanged: dual accumulators (accP/accN, final accP-accN store),
//    scalar sign branch (EXEC all-1s around WMMA), conflict-free LDS tiles,
//    global_prefetch_b8 of the next feature chunk.
// ---------------------------------------------------------------------------

typedef __attribute__((ext_vector_type(2))) float v2f;
typedef __attribute__((ext_vector_type(8))) float v8f;

#define NBLADES 8
#define BATCH   128
#define INFEAT  128
#define UNITS   64
#define CHUNK   32
#define XPAD    36   // 32 + 4; row stride ≡ 36 (mod 64 banks) -> conflict-free A loads

// Blade index <-> basis bitmask, BLADES order: (),e0,e1,e2,e01,e02,e12,e012.
// The permutation (swap 3<->4) is self-inverse; nibble i of 0x76534210 is
// perm(i), serving both idx->mask and mask->idx.
#define BLADE_PERM_TAB 0x76534210u
__device__ __forceinline__ int blade_perm(int i) {
  return (int)((BLADE_PERM_TAB >> (4 * i)) & 7u);
}

__global__ __launch_bounds__(256) void ga_mvprod_wmma(
    const float* __restrict__ x,   // (8, 128, 128)
    const float* __restrict__ w,   // (8, 128, 64)
    float* __restrict__ out)       // (8, 128, 64)
{
  __shared__ float xs[NBLADES][16][XPAD];   // 18.0 KB
  __shared__ float ws[NBLADES][CHUNK][16];  // 16.0 KB

  const int tid   = threadIdx.x;
  const int lane  = tid & 31;       // wave32
  // Wave id is uniform; force into an SGPR so all Cayley math and the sign
  // branches are scalar (WMMA requires EXEC all-1s).
  const int o     = __builtin_amdgcn_readfirstlane(tid >> 5);  // output blade
  const int mtile = blockIdx.x >> 2;
  const int ntile = blockIdx.x & 3;

  const int mo = blade_perm(o);     // output blade mask (SGPR, from immediate)

  // Precompute this wave's Cayley row: for each ia, the w-blade index ib and
  // the reorder sign. ma is a literal after unrolling, so this is a short
  // pure-SALU prologue; results packed into two scalar words.
  //   neg(am,bm) = (bm0 & (am1^am2)) ^ (bm1 & am2)   (positive signature)
  unsigned ibword = 0, negword = 0;
#pragma unroll
  for (int ia = 0; ia < NBLADES; ++ia) {
    const int ma  = blade_perm(ia);           // literal
    const int mb  = mo ^ ma;                  // product blade mask (SGPR)
    const int ib  = blade_perm(mb);           // w blade index
    const int a1  = (ma >> 1) & 1, a2 = (ma >> 2) & 1;  // literals
    const int neg = ((mb & (a1 ^ a2)) ^ ((mb >> 1) & a2)) & 1;
    ibword  |= (unsigned)ib  << (3 * ia);
    negword |= (unsigned)neg << ia;
  }

  // WMMA fragment coordinates (per ISA 32-bit A/B/C layouts, wave32)
  const int fm = lane & 15;         // A: row M / B,C: col N
  const int kb = (lane >> 4) * 2;   // half-wave K offset {0,2}

  v8f accP = {};                    // positive-sign pairs
  v8f accN = {};                    // negative-sign pairs

  for (int ch = 0; ch < INFEAT / CHUNK; ++ch) {
    __syncthreads();  // protect previous chunk's LDS reads

    // ---- stage x chunk: 8 blades x 16 rows x 32 feats, float4 per task ----
#pragma unroll
    for (int r = 0; r < 4; ++r) {
      int idx = tid + 256 * r;          // 0..1023
      int ia  = idx >> 7;               // 128 float4 tasks per blade
      int rem = idx & 127;
      int m   = rem >> 3;               // row in tile
      int g   = rem & 7;                // float4 group (8 per 32 feats)
      float4 v = *(const float4*)(x + ((size_t)(ia * BATCH + mtile * 16 + m) * INFEAT
                                       + ch * CHUNK + 4 * g));
      *(float4*)(&xs[ia][m][4 * g]) = v;
    }
    // ---- stage w chunk: 8 blades x 32 k x 16 cols ----
#pragma unroll
    for (int r = 0; r < 4; ++r) {
      int idx = tid + 256 * r;
      int ib  = idx >> 7;
      int rem = idx & 127;
      int k   = rem >> 2;               // 32 k-rows
      int g   = rem & 3;                // 4 float4 groups per row
      float4 v = *(const float4*)(w + ((size_t)(ib * INFEAT + ch * CHUNK + k) * UNITS
                                       + ntile * 16 + 4 * g));
      *(float4*)(&ws[ib][k][4 * g]) = v;
    }
    __syncthreads();

    // ---- prefetch next chunk while this one computes (global_prefetch_b8).
    // x: one lane per (blade, row) -> each 128B cacheline = one full chunk row.
    // w: one lane per (blade, k)   -> covers the 64B of this ntile's columns.
    if (ch + 1 < INFEAT / CHUNK) {
      __builtin_prefetch(x + ((size_t)((tid >> 5) * BATCH + mtile * 16 + (tid & 15)) * INFEAT
                              + (ch + 1) * CHUNK), 0, 3);
      __builtin_prefetch(w + ((size_t)((tid >> 5) * INFEAT + (ch + 1) * CHUNK + (tid & 31)) * UNITS
                              + ntile * 16), 0, 3);
    }

    // ---- accumulate: 8 blade-pairs x 8 K-steps of v_wmma_f32_16x16x4_f32.
    // ib/neg come from packed scalar words; sign handled by accumulator
    // selection behind a scalar branch (full EXEC).
    for (int ia = 0; ia < NBLADES; ++ia) {
      const int ib = (int)((ibword >> (3 * ia)) & 7u);
      if ((negword >> ia) & 1u) {
#pragma unroll
        for (int k = 0; k < CHUNK; k += 4) {
          v2f a, b;
          a.x = xs[ia][fm][k + kb];      // A 16x4: lanes 0-15 K={0,1}, 16-31 K={2,3}
          a.y = xs[ia][fm][k + kb + 1];
          b.x = ws[ib][k + kb][fm];      // B 4x16: mirrored, N = lane&15
          b.y = ws[ib][k + kb + 1][fm];
          accN = __builtin_amdgcn_wmma_f32_16x16x4_f32(
              false, a, false, b, (short)0, accN, false, false);
        }
      } else {
#pragma unroll
        for (int k = 0; k < CHUNK; k += 4) {
          v2f a, b;
          a.x = xs[ia][fm][k + kb];
          a.y = xs[ia][fm][k + kb + 1];
          b.x = ws[ib][k + kb][fm];
          b.y = ws[ib][k + kb + 1][fm];
          accP = __builtin_amdgcn_wmma_f32_16x16x4_f32(
              false, a, false, b, (short)0, accP, false, false);
        }
      }
    }
  }

  // ---- store C/D: VGPR r -> M=r (lanes 0-15) or M=8+r (lanes 16-31) ----
  const int n     = lane & 15;
  const int mbase = (lane >> 4) * 8;
#pragma unroll
  for (int r = 0; r < 8; ++r) {
    out[((size_t)o * BATCH + mtile * 16 + mbase + r) * UNITS + ntile * 16 + n] =
        accP[r] - accN[r];
  }
}

extern "C" void kernel_launch(void* const* d_in, const int* in_sizes, int n_in,
                              void* d_out, int out_size, void* d_ws, size_t ws_size,
                              hipStream_t stream) {
  (void)in_sizes; (void)n_in; (void)out_size; (void)d_ws; (void)ws_size;
  const float* x = (const float*)d_in[0];   // (8,128,128) f32
  const float* w = (const float*)d_in[1];   // (8,128,64)  f32
  float* out = (float*)d_out;               // (8,128,64)  f32
  dim3 grid(32);    // 8 mtiles x 4 ntiles
  dim3 block(256);  // 8 wave32 waves = 8 output blades
  hipLaunchKernelGGL(ga_mvprod_wmma, grid, block, 0, stream, x, w, out);
}